// RNNEncoder_39101382262798
// MI455X (gfx1250) — compile-verified
//
#include <hip/hip_runtime.h>

typedef __attribute__((ext_vector_type(2))) float v2f;
typedef __attribute__((ext_vector_type(8))) float v8f;

#define NN   1024
#define FF   32
#define DD   16
#define BB   4

// ---------------------------------------------------------------------------
// Step 0: m_prev == 0  ->  msg[n,d,:] = w_node[n, nb[n,d]] * x[n,:]
// One wave per (b,n); lane = feature.
// ---------------------------------------------------------------------------
__global__ void k_step0(const long long* __restrict__ nb,
                        const float* __restrict__ x,
                        const float* __restrict__ w_node,
                        float* __restrict__ msg)
{
    int wid  = blockIdx.x * (blockDim.x >> 5) + (threadIdx.x >> 5); // (b*N + n)
    int lane = threadIdx.x & 31;
    int b = wid >> 10;
    int n = wid & (NN - 1);
    const long long* nbb = nb + (size_t)b * NN * DD;
    float* mb = msg + (size_t)b * NN * DD * FF;
    float xv = x[((size_t)b * NN + n) * FF + lane];
#pragma unroll
    for (int d = 0; d < DD; ++d) {
        int s = (int)nbb[n * DD + d];
        mb[(n * DD + d) * FF + lane] = w_node[n * NN + s] * xv;
    }
}

// ---------------------------------------------------------------------------
// Generic step: compressed-form _compose_step.
//   gathered[n,d,:] = relu(msgOld[s, e, :]) where s=nb[n,d], e = last slot with
//                     nb[s,e]==n (reverse edge), else 0.
//   S[n,:]    = sum_d w_msg[n,s_d] * gathered[n,d,:]
//   msgNew[n,d,:] = w_node[n,s_d]*x[n,:] + S[n,:] - w_msg[n,s_d]*gathered[n,d,:]
// One wave per (b,n); lane = feature.
// ---------------------------------------------------------------------------
__global__ void k_step(const long long* __restrict__ nb,
                       const float* __restrict__ x,
                       const float* __restrict__ w_node,
                       const float* __restrict__ w_msg,
                       const float* __restrict__ msgOld,
                       float* __restrict__ msgNew)
{
    int wid  = blockIdx.x * (blockDim.x >> 5) + (threadIdx.x >> 5);
    int lane = threadIdx.x & 31;
    int b = wid >> 10;
    int n = wid & (NN - 1);
    const long long* nbb = nb + (size_t)b * NN * DD;
    const float* mo = msgOld + (size_t)b * NN * DD * FF;
    float* mn = msgNew + (size_t)b * NN * DD * FF;

    float S = 0.f;
    float g[DD], wm[DD], wn[DD];
#pragma unroll
    for (int d = 0; d < DD; ++d) {
        int s = (int)nbb[n * DD + d];
        int e = -1;
#pragma unroll
        for (int q = 0; q < DD; ++q)
            if ((int)nbb[s * DD + q] == n) e = q;   // last match = scatter last-wins
        float gv = 0.f;
        if (e >= 0) {
            gv = mo[(s * DD + e) * FF + lane];
            gv = gv > 0.f ? gv : 0.f;
        }
        float wmv = w_msg[n * NN + s];
        g[d] = gv; wm[d] = wmv; wn[d] = w_node[n * NN + s];
        S += wmv * gv;
    }
    float xv = x[((size_t)b * NN + n) * FF + lane];
#pragma unroll
    for (int d = 0; d < DD; ++d)
        mn[(n * DD + d) * FF + lane] = wn[d] * xv + S - wm[d] * g[d];
}

// ---------------------------------------------------------------------------
// agg[n,:] = sum over distinct columns (keep only last duplicate slot) of
//            relu(msgFinal[n,d,:]).  One wave per (b,n).
// ---------------------------------------------------------------------------
__global__ void k_agg(const long long* __restrict__ nb,
                      const float* __restrict__ msg,
                      float* __restrict__ agg)
{
    int wid  = blockIdx.x * (blockDim.x >> 5) + (threadIdx.x >> 5);
    int lane = threadIdx.x & 31;
    int b = wid >> 10;
    int n = wid & (NN - 1);
    const long long* nbb = nb + (size_t)b * NN * DD;
    const float* mb = msg + (size_t)b * NN * DD * FF;
    float a = 0.f;
#pragma unroll
    for (int d = 0; d < DD; ++d) {
        int s = (int)nbb[n * DD + d];
        bool dup = false;
#pragma unroll
        for (int q = 0; q < DD; ++q)
            if (q > d && (int)nbb[n * DD + q] == s) dup = true;
        if (!dup) {
            float v = mb[(n * DD + d) * FF + lane];
            a += v > 0.f ? v : 0.f;
        }
    }
    agg[((size_t)b * NN + n) * FF + lane] = a;
}

// ---------------------------------------------------------------------------
// enc = relu(u_node @ x + agg @ u_msg^T)  via V_WMMA_F32_16X16X4_F32.
// Block = 4 waves; each wave owns one 16-row tile and both 16-col tiles of F=32.
// A layout (16x4 f32): lane%16 = M row, (lane/16)*2 = K pair base, 2 VGPRs.
// B layout (4x16 f32): lane%16 = N col, same K striping.
// D layout (16x16 f32): vgpr v -> M = v + 8*(lane/16); lane%16 = N.
// ---------------------------------------------------------------------------
__global__ void k_enc(const float* __restrict__ x,
                      const float* __restrict__ u_node,
                      const float* __restrict__ u_msg,
                      const float* __restrict__ agg,
                      float* __restrict__ enc)
{
    int b    = blockIdx.y;
    int wave = threadIdx.x >> 5;
    int lane = threadIdx.x & 31;
    int r0   = (blockIdx.x * 4 + wave) * 16;
    int m    = lane & 15;
    int kb   = (lane >> 4) * 2;

    const float* xb = x   + (size_t)b * NN * FF;
    const float* ab = agg + (size_t)b * NN * FF;
    v8f acc0 = {}; v8f acc1 = {};

    // u_node @ x :  K = 1024
    for (int k = 0; k < NN; k += 4) {
        v2f a;
        a.x = u_node[(r0 + m) * NN + k + kb];
        a.y = u_node[(r0 + m) * NN + k + kb + 1];
        v2f b0, b1;
        b0.x = xb[(k + kb) * FF + m];
        b0.y = xb[(k + kb + 1) * FF + m];
        b1.x = xb[(k + kb) * FF + 16 + m];
        b1.y = xb[(k + kb + 1) * FF + 16 + m];
        acc0 = __builtin_amdgcn_wmma_f32_16x16x4_f32(false, a, false, b0,
                                                     (short)0, acc0, false, false);
        acc1 = __builtin_amdgcn_wmma_f32_16x16x4_f32(false, a, false, b1,
                                                     (short)0, acc1, false, false);
    }
    // agg @ u_msg^T :  K = 32,  B[k][f] = u_msg[f][k]
    for (int k = 0; k < FF; k += 4) {
        v2f a;
        a.x = ab[(r0 + m) * FF + k + kb];
        a.y = ab[(r0 + m) * FF + k + kb + 1];
        v2f b0, b1;
        b0.x = u_msg[m * FF + k + kb];
        b0.y = u_msg[m * FF + k + kb + 1];
        b1.x = u_msg[(16 + m) * FF + k + kb];
        b1.y = u_msg[(16 + m) * FF + k + kb + 1];
        acc0 = __builtin_amdgcn_wmma_f32_16x16x4_f32(false, a, false, b0,
                                                     (short)0, acc0, false, false);
        acc1 = __builtin_amdgcn_wmma_f32_16x16x4_f32(false, a, false, b1,
                                                     (short)0, acc1, false, false);
    }

    float* eb = enc + (size_t)b * NN * FF;
    int half = lane >> 4;
#pragma unroll
    for (int v = 0; v < 8; ++v) {
        int row = r0 + v + 8 * half;
        int col = lane & 15;
        float e0 = acc0[v]; e0 = e0 > 0.f ? e0 : 0.f;
        float e1 = acc1[v]; e1 = e1 > 0.f ? e1 : 0.f;
        eb[row * FF + col]      = e0;
        eb[row * FF + 16 + col] = e1;
    }
}

// ---------------------------------------------------------------------------
// out[b,o] = relu(enc_flat[b,:] . W[o,:] + bias[o]).  M=4 -> streaming dot:
// one block per o, W row read once (float4), reused across all 4 batch rows.
// ---------------------------------------------------------------------------
__global__ void k_out(const float* __restrict__ enc,
                      const float* __restrict__ W,
                      const float* __restrict__ bias,
                      float* __restrict__ out)
{
    const int NF = NN * FF;           // 32768
    int o   = blockIdx.x;
    int tid = threadIdx.x;
    const float4* W4 = (const float4*)(W + (size_t)o * NF);
    const float4* E4 = (const float4*)enc;
    float a0 = 0.f, a1 = 0.f, a2 = 0.f, a3 = 0.f;
    for (int i = tid; i < NF / 4; i += 256) {
        float4 w  = W4[i];
        float4 e0 = E4[i];
        float4 e1 = E4[(NF / 4) + i];
        float4 e2 = E4[2 * (NF / 4) + i];
        float4 e3 = E4[3 * (NF / 4) + i];
        a0 += w.x * e0.x + w.y * e0.y + w.z * e0.z + w.w * e0.w;
        a1 += w.x * e1.x + w.y * e1.y + w.z * e1.z + w.w * e1.w;
        a2 += w.x * e2.x + w.y * e2.y + w.z * e2.z + w.w * e2.w;
        a3 += w.x * e3.x + w.y * e3.y + w.z * e3.z + w.w * e3.w;
    }
#pragma unroll
    for (int off = 16; off > 0; off >>= 1) {
        a0 += __shfl_down(a0, off, 32);
        a1 += __shfl_down(a1, off, 32);
        a2 += __shfl_down(a2, off, 32);
        a3 += __shfl_down(a3, off, 32);
    }
    __shared__ float red[4][8];
    int wv = tid >> 5, lane = tid & 31;
    if (lane == 0) { red[0][wv] = a0; red[1][wv] = a1; red[2][wv] = a2; red[3][wv] = a3; }
    __syncthreads();
    if (tid < 4) {
        float s = 0.f;
#pragma unroll
        for (int w = 0; w < 8; ++w) s += red[tid][w];
        s += bias[o];
        out[tid * 256 + o] = s > 0.f ? s : 0.f;
    }
}

// ---------------------------------------------------------------------------
extern "C" void kernel_launch(void* const* d_in, const int* in_sizes, int n_in,
                              void* d_out, int out_size, void* d_ws, size_t ws_size,
                              hipStream_t stream) {
    const float*     x      = (const float*)d_in[0];      // (B,N,F)
    const long long* nb     = (const long long*)d_in[1];  // (B,N,D) int64
    const float*     w_node = (const float*)d_in[3];      // (N,N)
    const float*     w_msg  = (const float*)d_in[4];      // (N,N)
    const float*     u_node = (const float*)d_in[5];      // (N,N)
    const float*     u_msg  = (const float*)d_in[6];      // (F,F)
    const float*     W      = (const float*)d_in[7];      // (OUT, N*F)
    const float*     bias   = (const float*)d_in[8];      // (OUT,)
    float*           out    = (float*)d_out;

    const size_t msgElems = (size_t)BB * NN * DD * FF;    // 2M floats = 8 MB
    float* msgA = (float*)d_ws;
    float* msgB = msgA + msgElems;
    // msgA is dead after the last step (final state lives in msgB):
    // overlay agg and enc into the msgA region (0.5 MB each << 8 MB).
    float* agg  = msgA;
    float* enc  = msgA + (size_t)BB * NN * FF;

    const int grid_wn = (BB * NN * 32) / 256;             // one wave per (b,n)
    dim3 blk(256);

    k_step0<<<grid_wn, blk, 0, stream>>>(nb, x, w_node, msgA);
    k_step <<<grid_wn, blk, 0, stream>>>(nb, x, w_node, w_msg, msgA, msgB);
    k_step <<<grid_wn, blk, 0, stream>>>(nb, x, w_node, w_msg, msgB, msgA);
    k_step <<<grid_wn, blk, 0, stream>>>(nb, x, w_node, w_msg, msgA, msgB);
    k_agg  <<<grid_wn, blk, 0, stream>>>(nb, msgB, agg);

    dim3 egrid(NN / 64, BB);                              // 4 waves/block * 16 rows
    k_enc<<<egrid, 128, 0, stream>>>(x, u_node, u_msg, agg, enc);

    k_out<<<256, 256, 0, stream>>>(enc, W, bias, out);
}